// DeltaNet_22488448762193
// MI455X (gfx1250) — compile-verified
//
#include <hip/hip_runtime.h>
#include <hip/hip_bf16.h>

// DeltaNet-style gated linear attention, fp32 end-to-end on gfx1250.
// - All GEMMs use V_WMMA_F32_16X16X4_F32 (fp32 RNE, matches the reference).
// - Scan chunk tiles are staged LDS-side by the Tensor Data Mover
//   (tensor_load_to_lds + s_wait_tensorcnt), overlapped with gate math and
//   the score-phase WMMAs.

typedef __attribute__((ext_vector_type(2))) float v2f;
typedef __attribute__((ext_vector_type(8))) float v8f;
typedef unsigned int v4u __attribute__((ext_vector_type(4)));
typedef int v4i __attribute__((ext_vector_type(4)));
typedef int v8i __attribute__((ext_vector_type(8)));

#define BB 2
#define LL 2048
#define DD 1024
#define HH 8
#define DH 128
#define NN (BB * LL)
#define CH 64   // scan chunk length

__device__ __forceinline__ v8f wmma4(v2f a, v2f b, v8f c) {
  return __builtin_amdgcn_wmma_f32_16x16x4_f32(false, a, false, b, (short)0, c,
                                               false, false);
}

// TDM: DMA one 64x128 fp32 tile (row-major, contiguous) global -> LDS.
// D# built per CDNA5 ISA 8.3/8.4: group0 {count=1, lds_addr, gaddr57, type=2},
// group1 {data_size=4B, tensor 128x64 stride 128, tile 128x64}, groups2/3 = 0.
// This toolchain's builtin takes 6 args (g0,g1,g2,g3,g4,cpol).
__device__ __forceinline__ void tdm_load_tile_64x128(unsigned lds_addr,
                                                     const float* gp) {
  unsigned long long ga = (unsigned long long)(uintptr_t)gp;
  v4u g0 = {1u,                                   // count=1, no gather
            lds_addr,                             // lds_addr (bytes)
            (unsigned)ga,                         // global_addr[31:0]
            (unsigned)((ga >> 32) & 0x01ffffffu) | 0x80000000u}; // [56:32]|type2
  v8i g1 = {(int)(2u << 16),                      // data_size = 4 bytes
            (int)(128u << 16),                    // tensor_dim0 = 128
            (int)(64u << 16),                     // tensor_dim1 = 64
            (int)(128u << 16),                    // tile_dim0 = 128
            (int)64,                              // tile_dim1 = 64, tile_dim2=0
            (int)128,                             // tensor_dim0_stride = 128
            0, 0};
  v4i g2 = {0, 0, 0, 0};
  v4i g3 = {0, 0, 0, 0};
  v8i g4 = {0, 0, 0, 0, 0, 0, 0, 0};
  __builtin_amdgcn_tensor_load_to_lds(g0, g1, g2, g3, g4, 0);
}

// ---------------------------------------------------------------------------
// Kernel 1: q/k/v = x @ W^T with RoPE fused for q,k.  One 16x16 tile per
// wave, K=1024 (256 fp32 WMMAs).  Output layout [B*H, L, Dh].
// ---------------------------------------------------------------------------
__global__ void qkv_rope_kernel(const float* __restrict__ x,
                                const float* __restrict__ Wq,
                                const float* __restrict__ Wk,
                                const float* __restrict__ Wv,
                                float* __restrict__ qo,
                                float* __restrict__ ko,
                                float* __restrict__ vo) {
  const int wave = threadIdx.x >> 5;
  const int lane = threadIdx.x & 31;
  const int hl   = lane & 15;
  const int koff = (lane >> 4) * 2;
  const int voff = (lane >> 4) << 3;
  const int m0   = blockIdx.x * 16;
  const int col0 = blockIdx.y * 128 + wave * 16;
  const int z    = blockIdx.z;

  const float* W   = (z == 0) ? Wq : (z == 1) ? Wk : Wv;
  float*       out = (z == 0) ? qo : (z == 1) ? ko : vo;

  const float* arow = x + (size_t)(m0 + hl) * DD;
  const float* brow = W + (size_t)(col0 + hl) * DD;

  v8f acc = {};
  for (int kk = 0; kk < DD; kk += 4) {
    v2f a = *(const v2f*)(arow + kk + koff);
    v2f b = *(const v2f*)(brow + kk + koff);
    acc = wmma4(a, b, acc);
  }

  const int col = col0 + hl;
  const int h = col >> 7, dh = col & 127;
  float invf = 0.f;
  if (z < 2) invf = expf(-(float)(dh >> 1) * (9.210340371976184f / 64.f));
#pragma unroll
  for (int vv = 0; vv < 8; ++vv) {
    int m = m0 + vv + voff;
    int b_ = m >> 11, t = m & (LL - 1);
    float val = acc[vv];
    if (z < 2) {
      float other = __shfl_xor(val, 1, 32);   // RoPE pair lives in lane^1
      float ang = (float)t * invf, sn, cs;
      sincosf(ang, &sn, &cs);
      val = (dh & 1) ? (val * cs + other * sn) : (val * cs - other * sn);
    }
    out[((size_t)(((b_ << 3) + h) * LL + t)) * DH + dh] = val;
  }
}

// ---------------------------------------------------------------------------
// Kernel 2: per-(n,h) gates; one wave per (n,h), shuffle-reduced dots.
// ---------------------------------------------------------------------------
__global__ void gates_kernel(const float* __restrict__ x,
                             const float* __restrict__ Wbeta,
                             const float* __restrict__ bbeta,
                             const float* __restrict__ Walpha,
                             const float* __restrict__ balpha,
                             float* __restrict__ alpha,
                             float* __restrict__ beta) {
  const int g = blockIdx.x * 8 + (threadIdx.x >> 5);
  const int lane = threadIdx.x & 31;
  const int n = g >> 3, h = g & 7;
  const float* xr = x + (size_t)n * DD;
  const float* wa = Walpha + h * DD;
  const float* wb = Wbeta + h * DD;
  float aa = 0.f, bb = 0.f;
  for (int c = lane; c < DD; c += 32) {
    float xv = xr[c];
    aa += xv * wa[c];
    bb += xv * wb[c];
  }
#pragma unroll
  for (int off = 16; off > 0; off >>= 1) {
    aa += __shfl_xor(aa, off, 32);
    bb += __shfl_xor(bb, off, 32);
  }
  if (lane == 0) {
    int b = n >> 11, t = n & (LL - 1);
    size_t idx = (size_t)((b << 3) + h) * LL + t;
    alpha[idx] = 1.f / (1.f + expf(-(aa + balpha[h])));
    beta[idx]  = 0.85f + 0.149f * (1.f / (1.f + expf(-(bb + bbeta[h]))));
  }
}

// ---------------------------------------------------------------------------
// Kernel 3: chunked delta scan.  One workgroup per (b,h); state S^T resident
// in LDS.  Per 64-step chunk (P = beta prefix product):
//   qs = P_t q_t, ks = (alpha_s/P_s) k_s
//   A  = tril(qs ks^T);  O = qs S_prev + A v;  S = P_end (S + ks^T v)
// Chunk tiles DMA'd by TDM; q,k scaled in place after s_wait_tensorcnt(1),
// v released before phase 2 via s_wait_tensorcnt(0).
// ---------------------------------------------------------------------------
__global__ void __launch_bounds__(256, 1)
scan_kernel(const float* __restrict__ q, const float* __restrict__ k,
            const float* __restrict__ v, const float* __restrict__ alpha,
            const float* __restrict__ beta, float* __restrict__ o) {
  __shared__ float sST[DH * DH];   // 64 KB  state, TRANSPOSED: sST[e*128+d]
  __shared__ float sQ [CH * DH];   // 32 KB  scaled q (row-major [t][d])
  __shared__ float sK [CH * DH];   // 32 KB  scaled k (row-major [s][d])
  __shared__ float sKT[DH * CH];   // 32 KB  scaled k transposed [d][s]
  __shared__ float sV [CH * DH];   // 32 KB  v (row-major [s][e])
  __shared__ float sA [CH * CH];   // 16 KB  masked scores
  __shared__ float sP [CH];
  __shared__ float sAl[CH];
  __shared__ float sBe[CH];

  const int tid = threadIdx.x;
  const int wave = tid >> 5, lane = tid & 31;
  const int hl = lane & 15, koff = (lane >> 4) * 2, voff = (lane >> 4) << 3;
  const int bh = blockIdx.x;

  const float* qb = q + (size_t)bh * LL * DH;
  const float* kb = k + (size_t)bh * LL * DH;
  const float* vb = v + (size_t)bh * LL * DH;
  float*       ob = o + (size_t)bh * LL * DH;
  const float* ab  = alpha + (size_t)bh * LL;
  const float* bbp = beta + (size_t)bh * LL;

  const unsigned ldsQ = (unsigned)(uintptr_t)(void*)sQ;
  const unsigned ldsK = (unsigned)(uintptr_t)(void*)sK;
  const unsigned ldsV = (unsigned)(uintptr_t)(void*)sV;

  for (int i = tid; i < DH * DH; i += 256) sST[i] = 0.f;
  __syncthreads();

  for (int c0 = 0; c0 < LL; c0 += CH) {
    // --- kick off TDM DMAs for this chunk (wave 0 only; in-order q,k,v) ---
    if (wave == 0) {
      tdm_load_tile_64x128(ldsQ, qb + (size_t)c0 * DH);
      tdm_load_tile_64x128(ldsK, kb + (size_t)c0 * DH);
      tdm_load_tile_64x128(ldsV, vb + (size_t)c0 * DH);
    }

    // --- gates & decay prefix (overlapped with DMA) ---
    if (tid < CH) { sAl[tid] = ab[c0 + tid]; sBe[tid] = bbp[c0 + tid]; }
    __syncthreads();
    if (tid == 0) {
      float p = 1.f;
      for (int j = 0; j < CH; ++j) { p *= sBe[j]; sP[j] = p; }
    }
    if (wave == 0) __builtin_amdgcn_s_wait_tensorcnt(1);  // q,k landed
    __syncthreads();

    // --- scale q,k in place; build transposed k copy ---
    for (int i = tid; i < CH * DH; i += 256) {
      int t = i >> 7, d = i & 127;
      sQ[i] *= sP[t];
      float kv = sK[i] * (sAl[t] / sP[t]);
      sK[i] = kv;
      sKT[d * CH + t] = kv;
    }
    // warm L2 for the next chunk's DMAs
    if (c0 + CH < LL) {
      const size_t nb = (size_t)(c0 + CH) * DH + (size_t)tid * 32;
      __builtin_prefetch(qb + nb, 0, 1);
      __builtin_prefetch(kb + nb, 0, 1);
      __builtin_prefetch(vb + nb, 0, 1);
    }
    __syncthreads();

    // --- phase 1: A = tril(qs @ ks^T), 16 tiles (2 per wave), K=128 ---
    for (int ti = wave * 2; ti < wave * 2 + 2; ++ti) {
      int tm = ti >> 2, tn = ti & 3;
      v8f acc = {};
      const float* qa = &sQ[(tm * 16 + hl) * DH + koff];
      const float* kp = &sK[(tn * 16 + hl) * DH + koff];
      for (int kk = 0; kk < DH; kk += 4) {
        v2f a  = *(const v2f*)(qa + kk);
        v2f b2 = *(const v2f*)(kp + kk);
        acc = wmma4(a, b2, acc);
      }
      int s = tn * 16 + hl;
#pragma unroll
      for (int vv = 0; vv < 8; ++vv) {
        int t = tm * 16 + vv + voff;
        sA[t * CH + s] = (s <= t) ? acc[vv] : 0.f;
      }
    }
    if (wave == 0) __builtin_amdgcn_s_wait_tensorcnt(0);  // v landed
    __syncthreads();

    // --- phase 2: O = qs @ S_prev + A @ V, 32 tiles (4 per wave) ---
    for (int ti = wave * 4; ti < wave * 4 + 4; ++ti) {
      int tm = ti >> 3, te = ti & 7;
      int e = te * 16 + hl;
      v8f acc = {};
      const float* qa = &sQ[(tm * 16 + hl) * DH + koff];
      const float* sb = &sST[e * DH + koff];          // S^T: contiguous b64
      for (int kk = 0; kk < DH; kk += 4) {
        v2f a  = *(const v2f*)(qa + kk);
        v2f b2 = *(const v2f*)(sb + kk);
        acc = wmma4(a, b2, acc);
      }
      const float* ap = &sA[(tm * 16 + hl) * CH + koff];
      for (int kk = 0; kk < CH; kk += 4) {
        v2f a = *(const v2f*)(ap + kk);
        int kr = kk + koff;
        v2f b2 = {sV[kr * DH + e], sV[(kr + 1) * DH + e]};
        acc = wmma4(a, b2, acc);
      }
#pragma unroll
      for (int vv = 0; vv < 8; ++vv) {
        int t = tm * 16 + vv + voff;
        ob[(size_t)(c0 + t) * DH + e] = acc[vv];
      }
    }
    __syncthreads();

    // --- phase 3: S = Pend * (S + ks^T @ V); wave w owns d-rows 16w..16w+15 ---
    {
      float pend = sP[CH - 1];
      const int dbase = wave * 16;
      const float* ka = &sKT[(dbase + hl) * CH + koff];
      for (int te = 0; te < 8; ++te) {
        int e = te * 16 + hl;
        float* cb = &sST[e * DH + dbase + voff];   // 8 consecutive floats
        v8f acc;
#pragma unroll
        for (int vv = 0; vv < 8; ++vv) acc[vv] = cb[vv];
        for (int kk = 0; kk < CH; kk += 4) {
          int ks = kk + koff;
          v2f a  = *(const v2f*)(ka + kk);
          v2f b2 = {sV[ks * DH + e], sV[(ks + 1) * DH + e]};
          acc = wmma4(a, b2, acc);
        }
#pragma unroll
        for (int vv = 0; vv < 8; ++vv) cb[vv] = acc[vv] * pend;
      }
    }
    __syncthreads();
  }
}

// ---------------------------------------------------------------------------
// Kernel 4: out = o @ Wo^T + bo (o read through [B,H,L,Dh] permutation).
// ---------------------------------------------------------------------------
__global__ void outproj_kernel(const float* __restrict__ o,
                               const float* __restrict__ Wo,
                               const float* __restrict__ bo,
                               float* __restrict__ out) {
  const int wave = threadIdx.x >> 5, lane = threadIdx.x & 31;
  const int hl = lane & 15, koff = (lane >> 4) * 2, voff = (lane >> 4) << 3;
  const int m0 = blockIdx.x * 16;
  const int col0 = blockIdx.y * 128 + wave * 16;
  const int n = m0 + hl;
  const int b_ = n >> 11, t = n & (LL - 1);
  const float* brow = Wo + (size_t)(col0 + hl) * DD;

  v8f acc = {};
  for (int kk = 0; kk < DD; kk += 4) {
    int c = kk + koff;                   // even -> pair stays inside a head
    int h = c >> 7, dh = c & 127;
    const float* ap = &o[((size_t)((b_ << 3) + h) * LL + t) * DH + dh];
    v2f a  = {ap[0], ap[1]};
    v2f b2 = *(const v2f*)(brow + kk + koff);
    acc = wmma4(a, b2, acc);
  }
  const int col = col0 + hl;
  float bias = bo[col];
#pragma unroll
  for (int vv = 0; vv < 8; ++vv) {
    int m = m0 + vv + voff;
    out[(size_t)m * DD + col] = acc[vv] + bias;
  }
}

// ---------------------------------------------------------------------------
extern "C" void kernel_launch(void* const* d_in, const int* in_sizes, int n_in,
                              void* d_out, int out_size, void* d_ws,
                              size_t ws_size, hipStream_t stream) {
  const float* x      = (const float*)d_in[0];
  const float* Wq     = (const float*)d_in[1];
  const float* Wk     = (const float*)d_in[2];
  const float* Wv     = (const float*)d_in[3];
  const float* Wbeta  = (const float*)d_in[4];
  const float* bbeta  = (const float*)d_in[5];
  const float* Walpha = (const float*)d_in[6];
  const float* balpha = (const float*)d_in[7];
  const float* Wo     = (const float*)d_in[8];
  const float* bo     = (const float*)d_in[9];
  float* out = (float*)d_out;

  float* ws    = (float*)d_ws;
  float* q     = ws;                         // [B*H, L, Dh]
  float* k     = q + (size_t)NN * DD;
  float* v     = k + (size_t)NN * DD;
  float* o     = v + (size_t)NN * DD;
  float* alpha = o + (size_t)NN * DD;        // [B*H, L]
  float* beta  = alpha + (size_t)BB * HH * LL;

  qkv_rope_kernel<<<dim3(NN / 16, DD / 128, 3), 256, 0, stream>>>(
      x, Wq, Wk, Wv, q, k, v);
  gates_kernel<<<dim3((NN * HH) / 8), 256, 0, stream>>>(
      x, Wbeta, bbeta, Walpha, balpha, alpha, beta);
  scan_kernel<<<dim3(BB * HH), 256, 0, stream>>>(q, k, v, alpha, beta, o);
  outproj_kernel<<<dim3(NN / 16, DD / 128), 256, 0, stream>>>(o, Wo, bo, out);
}